// SAGE_22041772163420
// MI455X (gfx1250) — compile-verified
//
#include <hip/hip_runtime.h>

#define N_NODES 50000
#define N_EDGES 800000
#define N_TILES (N_NODES / 16)   // 3125, exact

typedef __attribute__((ext_vector_type(2))) float v2f;
typedef __attribute__((ext_vector_type(8))) float v8f;

// ---------------------------------------------------------------- utilities

__global__ void zero_f32(float* __restrict__ p, int n) {
    int i = blockIdx.x * blockDim.x + threadIdx.x;
    int stride = gridDim.x * blockDim.x;
    for (; i < n; i += stride) p[i] = 0.0f;
}

__global__ void degree_kernel(const int* __restrict__ dst, float* __restrict__ deg) {
    int e = blockIdx.x * blockDim.x + threadIdx.x;
    if (e < N_EDGES) unsafeAtomicAdd(&deg[dst[e]], 1.0f);
}

__global__ void invdeg_kernel(float* __restrict__ deg) {
    int i = blockIdx.x * blockDim.x + threadIdx.x;
    if (i < N_NODES) deg[i] = 1.0f / fmaxf(deg[i], 1.0f);
}

// agg[dst[e], :] += feat[src[e], :]; one thread per (edge, 4-float group).
// Feature table (12.8 MB) fits in the 192 MB L2, so gathers hit L2; the
// scatter uses native global_atomic_add_f32.
__global__ void scatter_kernel(const float* __restrict__ feat,
                               const int* __restrict__ src,
                               const int* __restrict__ dst,
                               float* __restrict__ agg) {
    int t = blockIdx.x * blockDim.x + threadIdx.x;   // < 800000*16 = 12.8M
    if (t >= N_EDGES * 16) return;
    int e = t >> 4;
    int g = (t & 15) << 2;
    const float4 v = *(const float4*)(feat + src[e] * 64 + g);
    float* base = agg + dst[e] * 64 + g;
    unsafeAtomicAdd(base + 0, v.x);
    unsafeAtomicAdd(base + 1, v.y);
    unsafeAtomicAdd(base + 2, v.z);
    unsafeAtomicAdd(base + 3, v.w);
}

// ---------------------------------------------------------------- WMMA GEMM
// Out[16-node tile, NOUT] = X_tile @ Ws^T + (Agg_tile * inv) @ Wn^T + bias
// using V_WMMA_F32_16X16X4_F32, one wave per tile, K = 64 in steps of 4.
//
// Lane layouts (ISA 7.12.2, 32-bit operands, wave32):
//   A 16x4:  lanes 0-15 -> M=lane, {VGPR0,VGPR1} = K{0,1};
//            lanes 16-31 -> M=lane-16, K{2,3}       (contiguous pair per lane)
//   B 4x16:  lanes 0-15 -> N=lane, rows K{0,1}; lanes 16-31 -> N, rows K{2,3}
//   C/D:     VGPR i: lanes 0-15 -> (M=i, N=lane); lanes 16-31 -> (M=i+8, N)
template <int NOUT, bool RELU>
__global__ void sage_gemm(const float* __restrict__ X,    // [N,64] self input
                          const float* __restrict__ Agg,  // [N,64] neighbor sums
                          const float* __restrict__ inv,  // [N] 1/max(deg,1)
                          const float* __restrict__ Ws,   // [NOUT,64]
                          const float* __restrict__ Wn,   // [NOUT,64]
                          const float* __restrict__ bias, // [NOUT]
                          float* __restrict__ Out)        // [N,NOUT]
{
    constexpr int NTILES = NOUT / 16;
    const int lane = threadIdx.x & 31;
    const int wave = threadIdx.x >> 5;
    const int tile = blockIdx.x * (blockDim.x >> 5) + wave;
    if (tile >= N_TILES) return;   // wave-uniform: EXEC stays all-ones

    const int m  = lane & 15;      // A row / B column / D column
    const int kh = lane >> 4;      // which K pair this lane holds
    const int node   = tile * 16 + m;
    const int rowoff = node * 64;
    const float scale = inv[node];

    v8f acc[NTILES] = {};

    #pragma unroll
    for (int k = 0; k < 64; k += 4) {
        const int kb = k + kh * 2;
        v2f a_s, a_n;
        a_s.x = X[rowoff + kb];
        a_s.y = X[rowoff + kb + 1];
        a_n.x = Agg[rowoff + kb] * scale;
        a_n.y = Agg[rowoff + kb + 1] * scale;
        #pragma unroll
        for (int nt = 0; nt < NTILES; ++nt) {
            const int w = (nt * 16 + m) * 64 + kb;   // W[n][k] = B[k][n]
            v2f b_s, b_n;
            b_s.x = Ws[w];
            b_s.y = Ws[w + 1];
            b_n.x = Wn[w];
            b_n.y = Wn[w + 1];
            acc[nt] = __builtin_amdgcn_wmma_f32_16x16x4_f32(
                false, a_s, false, b_s, (short)0, acc[nt], false, false);
            acc[nt] = __builtin_amdgcn_wmma_f32_16x16x4_f32(
                false, a_n, false, b_n, (short)0, acc[nt], false, false);
        }
    }

    #pragma unroll
    for (int nt = 0; nt < NTILES; ++nt) {
        const float bv = bias[nt * 16 + m];
        #pragma unroll
        for (int i = 0; i < 8; ++i) {
            const int M = i + kh * 8;
            float v = acc[nt][i] + bv;
            if (RELU) v = fmaxf(v, 0.0f);
            Out[(tile * 16 + M) * NOUT + nt * 16 + m] = v;
        }
    }
}

// ---------------------------------------------------------------- launch

extern "C" void kernel_launch(void* const* d_in, const int* in_sizes, int n_in,
                              void* d_out, int out_size, void* d_ws, size_t ws_size,
                              hipStream_t stream) {
    const float* x   = (const float*)d_in[0];
    const int*   src = (const int*)d_in[1];
    const int*   dst = (const int*)d_in[2];
    const float* W1s = (const float*)d_in[3];
    const float* W1n = (const float*)d_in[4];
    const float* b1  = (const float*)d_in[5];
    const float* W2s = (const float*)d_in[6];
    const float* W2n = (const float*)d_in[7];
    const float* b2  = (const float*)d_in[8];
    float* out = (float*)d_out;

    // workspace: deg[N] | agg[N*64] | h1[N*64]   (~25.8 MB of f32)
    float* deg = (float*)d_ws;
    float* agg = deg + N_NODES;
    float* h1  = agg + N_NODES * 64;

    const int scatter_blocks = (N_EDGES * 16 + 255) / 256;
    const int gemm_blocks    = (N_TILES + 3) / 4;       // 4 waves per block

    // ---- layer 1
    zero_f32<<<2048, 256, 0, stream>>>(deg, N_NODES * 65);   // deg + agg contiguous
    degree_kernel<<<(N_EDGES + 255) / 256, 256, 0, stream>>>(dst, deg);
    invdeg_kernel<<<(N_NODES + 255) / 256, 256, 0, stream>>>(deg);
    scatter_kernel<<<scatter_blocks, 256, 0, stream>>>(x, src, dst, agg);
    sage_gemm<64, true><<<gemm_blocks, 128, 0, stream>>>(x, agg, deg, W1s, W1n, b1, h1);

    // ---- layer 2
    zero_f32<<<2048, 256, 0, stream>>>(agg, N_NODES * 64);
    scatter_kernel<<<scatter_blocks, 256, 0, stream>>>(h1, src, dst, agg);
    sage_gemm<32, false><<<gemm_blocks, 128, 0, stream>>>(h1, agg, deg, W2s, W2n, b2, out);
}